// BatchedICP_20486994002017
// MI455X (gfx1250) — compile-verified
//
#include <hip/hip_runtime.h>
#include <math.h>

// ---------------- problem constants (from reference) ----------------
#define BATCH 16
#define NPTS  2048
#define NITER 10

#define TILES (NPTS / 16)            // 128 column tiles of targets
#define ROWS_PER_WG 256              // source rows handled per workgroup
#define WGS_PER_BATCH (NPTS / ROWS_PER_WG)   // 8
#define TA 256                       // threads per block (8 waves of 32)
#define WAVES (TA / 32)
// each wave owns 2 row tiles (32 source rows): 256 rows / 8 waves / 16

typedef __attribute__((ext_vector_type(2))) float v2f;
typedef __attribute__((ext_vector_type(8))) float v8f;

// ---------------------------------------------------------------
// Kernel 0: initialize working state (must run every launch; ws is
// not re-poisoned/restored between graph replays).
// ---------------------------------------------------------------
__global__ __launch_bounds__(256) void icp_init_kernel(
    const float* __restrict__ src, float* __restrict__ cur,
    float* __restrict__ Rtot, float* __restrict__ ttot)
{
    const int i = blockIdx.x * blockDim.x + threadIdx.x;
    if (i < BATCH * NPTS * 3) cur[i] = src[i];
    if (i < BATCH * 9)        Rtot[i] = (((i % 9) % 4) == 0) ? 1.0f : 0.0f; // identity
    if (i < BATCH * 3)        ttot[i] = 0.0f;
}

// ---------------------------------------------------------------
// Kernel A: nearest neighbor via V_WMMA_F32_16X16X4_F32 + partial
// Kabsch sums.  score(s,t) = s.t - 0.5|t|^2  (argmax == NN argmin).
// A row (16x4):  (sx, sy, sz, 1)
// B col (4x16):  (tx, ty, tz, -0.5|t|^2)
// One ds_load_b64 feeds two WMMAs (two row tiles share the B tile),
// and the B load is software-pipelined one tile ahead.
// ---------------------------------------------------------------
__global__ __launch_bounds__(TA) void icp_nn_kernel(
    const float* __restrict__ cur, const float* __restrict__ tgt,
    float* __restrict__ partials)
{
    // B-matrix staged in exact WMMA layout: per tile, lane l holds
    //   v0 = (l<16 ? K0 : K2), v1 = (l<16 ? K1 : K3) at column N = l&15
    __shared__ float Bmat[TILES][32][2];      // 32 KB
    __shared__ float tp[NPTS][3];             // 24 KB raw targets for gather
    __shared__ float red[WAVES][16];          // deterministic reduction

    const int b  = blockIdx.x / WGS_PER_BATCH;
    const int wg = blockIdx.x % WGS_PER_BATCH;
    const float* tg = tgt + (size_t)b * NPTS * 3;
    const float* cu = cur + ((size_t)b * NPTS + (size_t)wg * ROWS_PER_WG) * 3;

    for (int j = threadIdx.x; j < NPTS; j += TA) {
        const float tx = tg[3*j+0], ty = tg[3*j+1], tz = tg[3*j+2];
        const float w  = -0.5f * (tx*tx + ty*ty + tz*tz);
        const int tile = j >> 4, n = j & 15;
        Bmat[tile][n][0]      = tx;  // K=0
        Bmat[tile][n][1]      = ty;  // K=1
        Bmat[tile][n + 16][0] = tz;  // K=2
        Bmat[tile][n + 16][1] = w;   // K=3
        tp[j][0] = tx; tp[j][1] = ty; tp[j][2] = tz;
    }
    __syncthreads();

    const int wave = threadIdx.x >> 5;
    const int lane = threadIdx.x & 31;
    const int half = lane >> 4;
    const int l16  = lane & 15;

    // Two row tiles per wave; load both A operands up front.
    const int rowBase0 = wave * 32;        // rows [rowBase0, rowBase0+15]
    const int rowBase1 = rowBase0 + 16;    // rows [rowBase1, rowBase1+15]

    const float p0x = cu[(rowBase0 + l16)*3 + 0];
    const float p0y = cu[(rowBase0 + l16)*3 + 1];
    const float p0z = cu[(rowBase0 + l16)*3 + 2];
    const float p1x = cu[(rowBase1 + l16)*3 + 0];
    const float p1y = cu[(rowBase1 + l16)*3 + 1];
    const float p1z = cu[(rowBase1 + l16)*3 + 2];

    // A layout (16x4 f32): lanes 0-15 -> (K0,K1), lanes 16-31 -> (K2,K3)
    v2f a0, a1;
    a0.x = half ? p0z : p0x;  a0.y = half ? 1.0f : p0y;
    a1.x = half ? p1z : p1x;  a1.y = half ? 1.0f : p1y;

    float bV0[8], bV1[8];
    int   bI0[8], bI1[8];
    #pragma unroll
    for (int r = 0; r < 8; ++r) {
        bV0[r] = -3.4e38f; bI0[r] = 0;
        bV1[r] = -3.4e38f; bI1[r] = 0;
    }

    // Software-pipelined B-tile load (branch-free rotation).
    v2f bb = *reinterpret_cast<const v2f*>(&Bmat[0][lane][0]);
    for (int t = 0; t < TILES; ++t) {
        const v2f bcur = bb;
        bb = *reinterpret_cast<const v2f*>(&Bmat[(t + 1) & (TILES - 1)][lane][0]);

        v8f c0 = {0.f,0.f,0.f,0.f,0.f,0.f,0.f,0.f};
        v8f c1 = {0.f,0.f,0.f,0.f,0.f,0.f,0.f,0.f};
        c0 = __builtin_amdgcn_wmma_f32_16x16x4_f32(
                 false, a0, false, bcur, (short)0, c0, false, false);
        c1 = __builtin_amdgcn_wmma_f32_16x16x4_f32(
                 false, a1, false, bcur, (short)0, c1, false, false);

        const int col = t * 16 + l16;
        #pragma unroll
        for (int r = 0; r < 8; ++r) {
            const bool g0 = c0[r] > bV0[r];
            bI0[r] = g0 ? col   : bI0[r];
            bV0[r] = g0 ? c0[r] : bV0[r];
            const bool g1 = c1[r] > bV1[r];
            bI1[r] = g1 ? col   : bI1[r];
            bV1[r] = g1 ? c1[r] : bV1[r];
        }
    }

    // Branch-free argmax across the 16 lanes of each half.
    #pragma unroll
    for (int m = 1; m < 16; m <<= 1) {
        #pragma unroll
        for (int r = 0; r < 8; ++r) {
            {
                const float ov = __shfl_xor(bV0[r], m, 32);
                const int   oi = __shfl_xor(bI0[r], m, 32);
                const bool take = (ov > bV0[r]) | ((ov == bV0[r]) & (oi < bI0[r]));
                bV0[r] = take ? ov : bV0[r];
                bI0[r] = take ? oi : bI0[r];
            }
            {
                const float ov = __shfl_xor(bV1[r], m, 32);
                const int   oi = __shfl_xor(bI1[r], m, 32);
                const bool take = (ov > bV1[r]) | ((ov == bV1[r]) & (oi < bI1[r]));
                bV1[r] = take ? ov : bV1[r];
                bI1[r] = take ? oi : bI1[r];
            }
        }
    }

    float sc[3] = {0,0,0};                     // sum of cur points
    float sq[3] = {0,0,0};                     // sum of matched targets
    float hh[9] = {0,0,0,0,0,0,0,0,0};         // sum cur_d * corr_e

    // lanes 0-7 -> rows 0-7, lanes 16-23 -> rows 8-15 of each tile
    if (l16 < 8) {
        int idx0 = 0, idx1 = 0;
        #pragma unroll
        for (int r = 0; r < 8; ++r) {
            if (l16 == r) { idx0 = bI0[r]; idx1 = bI1[r]; }
        }
        {
            const int row = rowBase0 + half * 8 + l16;
            const float cx = cu[row*3+0], cy = cu[row*3+1], cz = cu[row*3+2];
            const float qx = tp[idx0][0], qy = tp[idx0][1], qz = tp[idx0][2];
            sc[0] += cx; sc[1] += cy; sc[2] += cz;
            sq[0] += qx; sq[1] += qy; sq[2] += qz;
            hh[0] += cx*qx; hh[1] += cx*qy; hh[2] += cx*qz;
            hh[3] += cy*qx; hh[4] += cy*qy; hh[5] += cy*qz;
            hh[6] += cz*qx; hh[7] += cz*qy; hh[8] += cz*qz;
        }
        {
            const int row = rowBase1 + half * 8 + l16;
            const float cx = cu[row*3+0], cy = cu[row*3+1], cz = cu[row*3+2];
            const float qx = tp[idx1][0], qy = tp[idx1][1], qz = tp[idx1][2];
            sc[0] += cx; sc[1] += cy; sc[2] += cz;
            sq[0] += qx; sq[1] += qy; sq[2] += qz;
            hh[0] += cx*qx; hh[1] += cx*qy; hh[2] += cx*qz;
            hh[3] += cy*qx; hh[4] += cy*qy; hh[5] += cy*qz;
            hh[6] += cz*qx; hh[7] += cz*qy; hh[8] += cz*qz;
        }
    }

    // Deterministic wave reduction (fixed xor tree), then fixed-order sum.
    float vals[16];
    vals[0]=sc[0]; vals[1]=sc[1]; vals[2]=sc[2];
    vals[3]=sq[0]; vals[4]=sq[1]; vals[5]=sq[2];
    #pragma unroll
    for (int k = 0; k < 9; ++k) vals[6+k] = hh[k];
    vals[15] = 0.0f;
    #pragma unroll
    for (int m = 1; m < 32; m <<= 1) {
        #pragma unroll
        for (int k = 0; k < 16; ++k) vals[k] += __shfl_xor(vals[k], m, 32);
    }
    if (lane == 0) {
        #pragma unroll
        for (int k = 0; k < 16; ++k) red[wave][k] = vals[k];
    }
    __syncthreads();
    if (threadIdx.x < 16) {
        float s = 0.0f;
        for (int w = 0; w < WAVES; ++w) s += red[w][threadIdx.x];
        partials[(size_t)(b * WGS_PER_BATCH + wg) * 16 + threadIdx.x] = s;
    }
}

// ---------------------------------------------------------------
// Kernel B: per-batch Kabsch (3x3 SVD via Jacobi on H^T H), update
// accumulated transform, apply rigid transform to all points.
// ---------------------------------------------------------------
__device__ inline float det3(const float M[3][3]) {
    return M[0][0]*(M[1][1]*M[2][2]-M[1][2]*M[2][1])
         - M[0][1]*(M[1][0]*M[2][2]-M[1][2]*M[2][0])
         + M[0][2]*(M[1][0]*M[2][1]-M[1][1]*M[2][0]);
}

__global__ __launch_bounds__(256) void icp_update_kernel(
    float* __restrict__ cur, const float* __restrict__ partials,
    float* __restrict__ Rtot, float* __restrict__ ttot)
{
    __shared__ float Rs[9], ts[3];
    const int b = blockIdx.x;

    if (threadIdx.x == 0) {
        float p[15];
        for (int k = 0; k < 15; ++k) p[k] = 0.0f;
        for (int w = 0; w < WGS_PER_BATCH; ++w)
            for (int k = 0; k < 15; ++k)
                p[k] += partials[(size_t)(b * WGS_PER_BATCH + w) * 16 + k];

        const float invN = 1.0f / (float)NPTS;
        const float cs[3] = {p[0]*invN, p[1]*invN, p[2]*invN};
        const float ct[3] = {p[3]*invN, p[4]*invN, p[5]*invN};
        float H[3][3];
        for (int i = 0; i < 3; ++i)
            for (int j = 0; j < 3; ++j)
                H[i][j] = p[6 + 3*i + j] - (float)NPTS * cs[i] * ct[j];

        // Jacobi eigendecomposition of S = H^T H  ->  V, eigenvalues
        float S[3][3], V[3][3];
        for (int i = 0; i < 3; ++i)
            for (int j = 0; j < 3; ++j) {
                S[i][j] = H[0][i]*H[0][j] + H[1][i]*H[1][j] + H[2][i]*H[2][j];
                V[i][j] = (i == j) ? 1.0f : 0.0f;
            }
        for (int sweep = 0; sweep < 10; ++sweep) {
            for (int pair = 0; pair < 3; ++pair) {
                const int pp = (pair == 2) ? 1 : 0;
                const int qq = (pair == 0) ? 1 : 2;
                const float apq = S[pp][qq];
                if (fabsf(apq) > 1e-20f) {
                    const float theta = (S[qq][qq] - S[pp][pp]) / (2.0f * apq);
                    const float t = ((theta >= 0.0f) ? 1.0f : -1.0f) /
                                    (fabsf(theta) + sqrtf(theta*theta + 1.0f));
                    const float c = rsqrtf(t*t + 1.0f);
                    const float s = t * c;
                    for (int k = 0; k < 3; ++k) {
                        const float skp = S[k][pp], skq = S[k][qq];
                        S[k][pp] = c*skp - s*skq;
                        S[k][qq] = s*skp + c*skq;
                    }
                    for (int k = 0; k < 3; ++k) {
                        const float spk = S[pp][k], sqk = S[qq][k];
                        S[pp][k] = c*spk - s*sqk;
                        S[qq][k] = s*spk + c*sqk;
                    }
                    for (int k = 0; k < 3; ++k) {
                        const float vkp = V[k][pp], vkq = V[k][qq];
                        V[k][pp] = c*vkp - s*vkq;
                        V[k][qq] = s*vkp + c*vkq;
                    }
                }
            }
        }
        float lam[3] = {S[0][0], S[1][1], S[2][2]};
        for (int pass = 0; pass < 2; ++pass)
            for (int i = 0; i < 2; ++i)
                if (lam[i] < lam[i+1]) {
                    const float tl = lam[i]; lam[i] = lam[i+1]; lam[i+1] = tl;
                    for (int k = 0; k < 3; ++k) {
                        const float tv = V[k][i]; V[k][i] = V[k][i+1]; V[k][i+1] = tv;
                    }
                }
        // U columns = H v_i / sigma_i
        float U[3][3];
        for (int i = 0; i < 3; ++i) {
            float u0 = H[0][0]*V[0][i] + H[0][1]*V[1][i] + H[0][2]*V[2][i];
            float u1 = H[1][0]*V[0][i] + H[1][1]*V[1][i] + H[1][2]*V[2][i];
            float u2 = H[2][0]*V[0][i] + H[2][1]*V[1][i] + H[2][2]*V[2][i];
            const float nn = sqrtf(u0*u0 + u1*u1 + u2*u2);
            if (nn > 1e-12f) { const float inn = 1.0f/nn; u0*=inn; u1*=inn; u2*=inn; }
            U[0][i] = u0; U[1][i] = u1; U[2][i] = u2;
        }
        if (lam[2] < 1e-12f * (lam[0] + 1e-30f)) {  // degenerate sigma: complete basis
            U[0][2] = U[1][0]*U[2][1] - U[2][0]*U[1][1];
            U[1][2] = U[2][0]*U[0][1] - U[0][0]*U[2][1];
            U[2][2] = U[0][0]*U[1][1] - U[1][0]*U[0][1];
        }
        const float d = det3(U) * det3(V);       // det(V U^T)
        float R[3][3];
        for (int i = 0; i < 3; ++i)
            for (int j = 0; j < 3; ++j)
                R[i][j] = V[i][0]*U[j][0] + V[i][1]*U[j][1] + (V[i][2]*d)*U[j][2];
        float t[3];
        for (int i = 0; i < 3; ++i)
            t[i] = ct[i] - (R[i][0]*cs[0] + R[i][1]*cs[1] + R[i][2]*cs[2]);

        // accumulate totals: R_tot = R*R_tot ; t_tot = R*t_tot + t
        float Ro[9], to[3], Rn[9], tn[3];
        for (int k = 0; k < 9; ++k) Ro[k] = Rtot[(size_t)b*9 + k];
        for (int k = 0; k < 3; ++k) to[k] = ttot[(size_t)b*3 + k];
        for (int i = 0; i < 3; ++i)
            for (int j = 0; j < 3; ++j)
                Rn[3*i+j] = R[i][0]*Ro[0+j] + R[i][1]*Ro[3+j] + R[i][2]*Ro[6+j];
        for (int i = 0; i < 3; ++i)
            tn[i] = R[i][0]*to[0] + R[i][1]*to[1] + R[i][2]*to[2] + t[i];
        for (int k = 0; k < 9; ++k) Rtot[(size_t)b*9 + k] = Rn[k];
        for (int k = 0; k < 3; ++k) ttot[(size_t)b*3 + k] = tn[k];

        for (int i = 0; i < 3; ++i)
            for (int j = 0; j < 3; ++j) Rs[3*i+j] = R[i][j];
        ts[0] = t[0]; ts[1] = t[1]; ts[2] = t[2];
    }
    __syncthreads();

    const float R00=Rs[0],R01=Rs[1],R02=Rs[2],
                R10=Rs[3],R11=Rs[4],R12=Rs[5],
                R20=Rs[6],R21=Rs[7],R22=Rs[8];
    const float t0=ts[0], t1=ts[1], t2=ts[2];
    float* cb = cur + (size_t)b * NPTS * 3;
    for (int j = threadIdx.x; j < NPTS; j += blockDim.x) {
        const float x = cb[3*j+0], y = cb[3*j+1], z = cb[3*j+2];
        cb[3*j+0] = R00*x + R01*y + R02*z + t0;
        cb[3*j+1] = R10*x + R11*y + R12*z + t1;
        cb[3*j+2] = R20*x + R21*y + R22*z + t2;
    }
}

// ---------------------------------------------------------------
// Kernel C: pack outputs (aligned | R | t), flat in return order.
// ---------------------------------------------------------------
__global__ __launch_bounds__(256) void icp_final_kernel(
    const float* __restrict__ cur, const float* __restrict__ Rtot,
    const float* __restrict__ ttot, float* __restrict__ out)
{
    const int i = blockIdx.x * blockDim.x + threadIdx.x;
    const int nA = BATCH * NPTS * 3;
    const int nR = BATCH * 9;
    const int nT = BATCH * 3;
    if (i < nA)                 out[i] = cur[i];
    else if (i < nA + nR)       out[i] = Rtot[i - nA];
    else if (i < nA + nR + nT)  out[i] = ttot[i - nA - nR];
}

// ---------------------------------------------------------------
extern "C" void kernel_launch(void* const* d_in, const int* in_sizes, int n_in,
                              void* d_out, int out_size, void* d_ws, size_t ws_size,
                              hipStream_t stream) {
    (void)in_sizes; (void)n_in; (void)out_size; (void)ws_size;
    const float* src = (const float*)d_in[0];
    const float* tgt = (const float*)d_in[1];
    float* out = (float*)d_out;

    float* ws       = (float*)d_ws;
    float* cur      = ws;                                   // BATCH*NPTS*3
    float* partials = cur + (size_t)BATCH * NPTS * 3;       // BATCH*WGS_PER_BATCH*16
    float* Rtot     = partials + (size_t)BATCH * WGS_PER_BATCH * 16;  // BATCH*9
    float* ttot     = Rtot + (size_t)BATCH * 9;             // BATCH*3

    const int initN = BATCH * NPTS * 3;
    icp_init_kernel<<<(initN + 255) / 256, 256, 0, stream>>>(src, cur, Rtot, ttot);

    for (int it = 0; it < NITER; ++it) {
        icp_nn_kernel<<<BATCH * WGS_PER_BATCH, TA, 0, stream>>>(cur, tgt, partials);
        icp_update_kernel<<<BATCH, 256, 0, stream>>>(cur, partials, Rtot, ttot);
    }

    const int outN = BATCH * NPTS * 3 + BATCH * 9 + BATCH * 3;
    icp_final_kernel<<<(outN + 255) / 256, 256, 0, stream>>>(cur, Rtot, ttot, out);
}